// BuiltinGNN_20864951124556
// MI455X (gfx1250) — compile-verified
//
#include <hip/hip_runtime.h>
#include <hip/hip_bf16.h>

// ---------------------------------------------------------------------------
// GCN (3x GCNConv + mean-pool + linear head) for gfx1250.
// GEMMs use V_WMMA_F32_16X16X4_F32 (full fp32 accuracy; workload is
// L2-bandwidth bound, so fp32 WMMA costs nothing vs bf16 here).
// ---------------------------------------------------------------------------

#define NN 50000
#define EE 800000
#define HD 128      // DIN == H == 128
#define GG 64
#define OUTC 8

typedef __attribute__((ext_vector_type(2))) float v2f;
typedef __attribute__((ext_vector_type(8))) float v8f;

__device__ __forceinline__ void atomicAddF(float* p, float v) {
  unsafeAtomicAdd(p, v);   // lowers to global_atomic_add_f32 on CDNA5
}

// ---------------- degree / normalization ----------------
__global__ __launch_bounds__(256) void k_degree(const int* __restrict__ dst,
                                                float* __restrict__ deg, int E) {
  int e = blockIdx.x * 256 + threadIdx.x;
  if (e < E) atomicAddF(&deg[dst[e]], 1.0f);
}

__global__ __launch_bounds__(256) void k_dinv(const float* __restrict__ deg,
                                              float* __restrict__ dinv, int n) {
  int i = blockIdx.x * 256 + threadIdx.x;
  if (i < n) dinv[i] = __frsqrt_rn(deg[i] + 1.0f);   // deg incl. self-loop
}

// ---------------- WMMA fp32 GEMM: Hout[N,128] = X[N,128] @ W[128,128] -------
// grid.x = N/16 row strips; 256 threads = 8 waves; wave w owns cols [16w,16w+16)
__global__ __launch_bounds__(256) void k_gemm_wmma(const float* __restrict__ X,
                                                   const float* __restrict__ W,
                                                   float* __restrict__ Hout,
                                                   int n) {
  __shared__ float Xs[16 * HD];   // 8 KB strip of A
  const int rowBase = blockIdx.x * 16;

  // cooperative strip load (coalesced b32/b128 by compiler)
  for (int i = threadIdx.x; i < 16 * HD; i += 256) {
    int r = i >> 7, c = i & 127;
    Xs[i] = (rowBase + r < n) ? X[(size_t)(rowBase + r) * HD + c] : 0.0f;
  }
  __syncthreads();

  const int wave = threadIdx.x >> 5;          // 0..7 -> col tile
  const int lane = threadIdx.x & 31;
  const int colBase = wave * 16;
  const int mn = lane & 15;                   // M for A-frag, N for B/C-frag
  const int khalf = (lane >> 4) * 2;          // lanes 0-15: K {0,1}; 16-31: K {2,3}

  v8f c = {};
  #pragma unroll
  for (int k0 = 0; k0 < HD; k0 += 4) {
    v2f a, b;
    a.x = Xs[mn * HD + k0 + khalf];
    a.y = Xs[mn * HD + k0 + khalf + 1];
    b.x = W[(size_t)(k0 + khalf) * HD + colBase + mn];
    b.y = W[(size_t)(k0 + khalf + 1) * HD + colBase + mn];
    // D = A(16x4 f32) * B(4x16 f32) + C  -> v_wmma_f32_16x16x4_f32
    c = __builtin_amdgcn_wmma_f32_16x16x4_f32(false, a, false, b,
                                              (short)0, c, false, false);
  }

  const int mOff = (lane >> 4) * 8;           // C/D: vgpr r = row r (+8 hi lanes)
  #pragma unroll
  for (int r = 0; r < 8; ++r) {
    int row = rowBase + r + mOff;
    if (row < n) Hout[(size_t)row * HD + colBase + mn] = c[r];
  }
}

// ---------------- edge scatter: agg[dst] += coef * h[src] -------------------
// 8 edges per 256-thread block; 32 lanes/edge; float4 gather + 4 fp32 atomics
__global__ __launch_bounds__(256) void k_edge(const float* __restrict__ h,
                                              const int* __restrict__ src,
                                              const int* __restrict__ dst,
                                              const float* __restrict__ dinv,
                                              float* __restrict__ agg, int E) {
  int e = blockIdx.x * 8 + (threadIdx.x >> 5);
  if (e >= E) return;
  int lane = threadIdx.x & 31;
  int s = src[e], d = dst[e];
  float coef = dinv[s] * dinv[d];
  const float4* hp = (const float4*)(h + (size_t)s * HD);
  float4 v = hp[lane];                        // global_load_b128
  float* ap = agg + (size_t)d * HD + lane * 4;
  atomicAddF(ap + 0, v.x * coef);
  atomicAddF(ap + 1, v.y * coef);
  atomicAddF(ap + 2, v.z * coef);
  atomicAddF(ap + 3, v.w * coef);
}

// ---------------- combine: out = act?(agg + h*dinv^2 + b) -------------------
__global__ __launch_bounds__(256) void k_combine(const float* __restrict__ agg,
                                                 const float* __restrict__ hlin,
                                                 const float* __restrict__ dinv,
                                                 const float* __restrict__ bias,
                                                 float* __restrict__ out,
                                                 int n, int relu) {
  int gid = blockIdx.x * 256 + threadIdx.x;
  if (gid >= n * HD) return;
  int i = gid >> 7, f = gid & 127;
  float di = dinv[i];
  float v = agg[gid] + hlin[gid] * di * di + bias[f];
  out[gid] = relu ? fmaxf(v, 0.0f) : v;
}

// ---------------- mean pool ----------------
__global__ __launch_bounds__(128) void k_pool_sum(const float* __restrict__ act,
                                                  const int* __restrict__ batch,
                                                  float* __restrict__ sums, int n) {
  int i = blockIdx.x;
  if (i >= n) return;
  int g = batch[i];
  atomicAddF(&sums[(size_t)g * HD + threadIdx.x], act[(size_t)i * HD + threadIdx.x]);
}

__global__ __launch_bounds__(256) void k_pool_cnt(const int* __restrict__ batch,
                                                  float* __restrict__ cnt, int n) {
  int i = blockIdx.x * 256 + threadIdx.x;
  if (i < n) atomicAddF(&cnt[batch[i]], 1.0f);
}

// ---------------- head: out[64,8] = (sums/cnt) @ Wh + bh --------------------
__global__ __launch_bounds__(512) void k_head(const float* __restrict__ sums,
                                              const float* __restrict__ cnt,
                                              const float* __restrict__ Wh,
                                              const float* __restrict__ bh,
                                              float* __restrict__ out) {
  int g = threadIdx.x >> 3, o = threadIdx.x & 7;   // 64 x 8 = 512 threads
  float inv = 1.0f / fmaxf(cnt[g], 1.0f);
  float acc = bh[o];
  #pragma unroll 4
  for (int k = 0; k < HD; ++k)
    acc += sums[g * HD + k] * inv * Wh[k * OUTC + o];
  out[g * OUTC + o] = acc;
}

// ---------------------------------------------------------------------------
extern "C" void kernel_launch(void* const* d_in, const int* in_sizes, int n_in,
                              void* d_out, int out_size, void* d_ws, size_t ws_size,
                              hipStream_t stream) {
  const float* x     = (const float*)d_in[0];
  const int*   ei    = (const int*)  d_in[1];   // [2,E]: row0=src, row1=dst
  const int*   batch = (const int*)  d_in[2];
  const float* Wl[3] = {(const float*)d_in[3], (const float*)d_in[5], (const float*)d_in[7]};
  const float* bl[3] = {(const float*)d_in[4], (const float*)d_in[6], (const float*)d_in[8]};
  const float* Wh    = (const float*)d_in[9];
  const float* bh    = (const float*)d_in[10];
  float*       out   = (float*)d_out;

  const int n = in_sizes[0] / HD;        // 50000
  const int E = in_sizes[1] / 2;         // 800000
  const int* src = ei;
  const int* dst = ei + E;

  // workspace carve-up (~78 MB)
  const size_t NH = (size_t)n * HD;
  float* bufL   = (float*)d_ws;          // X@W (pre-norm)
  float* bufAgg = bufL   + NH;           // scatter accumulator
  float* bufAct = bufAgg + NH;           // layer output
  float* deg    = bufAct + NH;           // [n]
  float* dinv   = deg    + n;            // [n]
  float* sums   = dinv   + n;            // [G,H]
  float* cnt    = sums   + GG * HD;      // [G]

  // degree / normalization (shared by all layers)
  hipMemsetAsync(deg, 0, n * sizeof(float), stream);
  k_degree<<<(E + 255) / 256, 256, 0, stream>>>(dst, deg, E);
  k_dinv  <<<(n + 255) / 256, 256, 0, stream>>>(deg, dinv, n);

  const int rowTiles = (n + 15) / 16;    // 3125
  const float* cur = x;
  for (int l = 0; l < 3; ++l) {
    k_gemm_wmma<<<rowTiles, 256, 0, stream>>>(cur, Wl[l], bufL, n);
    hipMemsetAsync(bufAgg, 0, NH * sizeof(float), stream);
    k_edge<<<(E + 7) / 8, 256, 0, stream>>>(bufL, src, dst, dinv, bufAgg, E);
    k_combine<<<(int)((NH + 255) / 256), 256, 0, stream>>>(bufAgg, bufL, dinv,
                                                           bl[l], bufAct, n, l < 2 ? 1 : 0);
    cur = bufAct;
  }

  // global mean pool + head
  hipMemsetAsync(sums, 0, GG * HD * sizeof(float), stream);
  hipMemsetAsync(cnt,  0, GG * sizeof(float), stream);
  k_pool_sum<<<n, 128, 0, stream>>>(bufAct, batch, sums, n);
  k_pool_cnt<<<(n + 255) / 256, 256, 0, stream>>>(batch, cnt, n);
  k_head<<<1, 512, 0, stream>>>(sums, cnt, Wh, bh, out);
}